// ConvAttention_40046275067966
// MI455X (gfx1250) — compile-verified
//
#include <hip/hip_runtime.h>

// ---------------------------------------------------------------------------
// ConvAttention for MI455X (gfx1250, wave32)
//   pass 1: per-head 16x16 fp32 projections via V_WMMA_F32_16X16X4_F32
//   pass 2: zero output
//   pass 3: streaming edge pass, L2-resident gathers, run-length segment sum
// ---------------------------------------------------------------------------

typedef __attribute__((ext_vector_type(2))) float v2f;
typedef __attribute__((ext_vector_type(8))) float v8f;

// ---------------------------------------------------------------------------
// Pass 1: Q/K/V projections.
// Grid: one block per 16-node tile, block = (32,4): wave y handles head y.
// Per wave: 3 GEMMs (16x16x16 f32) = 12 x v_wmma_f32_16x16x4_f32.
//
// ISA layouts (CDNA5 ISA §7.12.2):
//   A (16x4 f32, 2 VGPRs): lane<16: row M=lane, {K, K+1}; lane>=16: row M=lane-16, {K+2, K+3}
//   B (4x16 f32, 2 VGPRs): mirrored: lane<16 carries rows {K, K+1} at col=lane,
//                          lane>=16 carries rows {K+2, K+3} at col=lane-16
//   C/D (16x16 f32, 8 VGPRs): vgpr v: lanes 0-15 -> M=v, lanes 16-31 -> M=v+8, N=lane%16
// ---------------------------------------------------------------------------
__global__ __launch_bounds__(128) void proj16_wmma(
    const float* __restrict__ x,        // (N, 64)
    const float* __restrict__ Wq,       // (4, 16, 16)
    const float* __restrict__ Wk,
    const float* __restrict__ Wv,
    float* __restrict__ Q,              // (N, 64)  e = h*16 + d
    float* __restrict__ Kt,
    float* __restrict__ Vt,
    int n_nodes)
{
    const int lane = threadIdx.x;        // 0..31
    const int head = threadIdx.y;        // 0..3
    const int row  = lane & 15;          // M (A) / N (B,D)
    const int hi   = lane >> 4;          // 0 | 1 (upper half-wave)
    const int base = blockIdx.x * 16;

    const float* Wqh = Wq + head * 256;
    const float* Wkh = Wk + head * 256;
    const float* Wvh = Wv + head * 256;

    int nodeA = base + row;
    if (nodeA >= n_nodes) nodeA = n_nodes - 1;       // clamp (reads only)
    const float* xrow = x + (long)nodeA * 64 + head * 16;

    v8f cq = {}; v8f ck = {}; v8f cv = {};

    #pragma unroll
    for (int s = 0; s < 4; ++s) {
        const int ka = 4 * s + 2 * hi;               // this lane's first K

        v2f a;                                       // A tile: x[node, head*16 + k]
        a.x = xrow[ka + 0];
        a.y = xrow[ka + 1];

        v2f bq, bk, bv;                              // B tile: W[head][k][e], e = row
        bq.x = Wqh[(ka + 0) * 16 + row];  bq.y = Wqh[(ka + 1) * 16 + row];
        bk.x = Wkh[(ka + 0) * 16 + row];  bk.y = Wkh[(ka + 1) * 16 + row];
        bv.x = Wvh[(ka + 0) * 16 + row];  bv.y = Wvh[(ka + 1) * 16 + row];

        cq = __builtin_amdgcn_wmma_f32_16x16x4_f32(false, a, false, bq, (short)0, cq, false, false);
        ck = __builtin_amdgcn_wmma_f32_16x16x4_f32(false, a, false, bk, (short)0, ck, false, false);
        cv = __builtin_amdgcn_wmma_f32_16x16x4_f32(false, a, false, bv, (short)0, cv, false, false);
    }

    // Store D. Fast path: whole tile in range (uniform test, branchless stores,
    // lets the compiler clause the 24 global_store_b32). Slow path: ragged tile.
    const int  moff  = 8 * hi;
    const long obase = (long)(base + moff) * 64 + head * 16 + row;   // row stride 64

    if (base + 16 <= n_nodes) {
        #pragma unroll
        for (int v = 0; v < 8; ++v) {
            const long o = obase + (long)v * 64;
            Q[o]  = cq[v];
            Kt[o] = ck[v];
            Vt[o] = cv[v];
        }
    } else {
        #pragma unroll
        for (int v = 0; v < 8; ++v) {
            if (base + moff + v < n_nodes) {
                const long o = obase + (long)v * 64;
                Q[o]  = cq[v];
                Kt[o] = ck[v];
                Vt[o] = cv[v];
            }
        }
    }
}

// ---------------------------------------------------------------------------
// Pass 2: zero the output accumulator (harness poisons d_out).
// ---------------------------------------------------------------------------
__global__ __launch_bounds__(256) void zero_f32(float* __restrict__ p, long n)
{
    long i = (long)blockIdx.x * blockDim.x + threadIdx.x;
    const long stride = (long)gridDim.x * blockDim.x;
    for (; i < n; i += stride) p[i] = 0.0f;
}

// ---------------------------------------------------------------------------
// Pass 3: edge pass.
// Each wave owns PPW consecutive pairs. Lane owns channels e = 2*lane, 2*lane+1
// (both inside head h = lane/8). Per-head dot = 8-lane shfl_xor reduction.
// idx_i is sorted -> register accumulation per run, atomic flush on change.
// ---------------------------------------------------------------------------
__global__ __launch_bounds__(256) void edge_attn(
    const float* __restrict__ Q,
    const float* __restrict__ Kt,
    const float* __restrict__ Vt,
    const float* __restrict__ w_ij,     // (P, 64) streamed from HBM
    const float* __restrict__ phi,      // (P,)
    const float* __restrict__ pmask,    // (P,)
    const int*   __restrict__ idx_i,    // (P,) sorted ascending
    const int*   __restrict__ idx_j,    // (P,)
    float* __restrict__ out,            // (N, 64)
    long n_pairs, int ppw)
{
    const int lane = threadIdx.x & 31;
    const int wid  = blockIdx.x * (blockDim.x >> 5) + (threadIdx.x >> 5);

    long p0 = (long)wid * ppw;
    if (p0 >= n_pairs) return;
    long p1 = p0 + ppw;
    if (p1 > n_pairs) p1 = n_pairs;

    const int e = lane << 1;            // channel pair base (0..62)

    float acc0 = 0.0f, acc1 = 0.0f;
    int cur = idx_i[p0];

    for (long p = p0; p < p1; ++p) {
        const int i = idx_i[p];
        const int j = idx_j[p];

        if (i != cur) {                 // wave-uniform flush
            float* o = out + (long)cur * 64 + e;
            __hip_atomic_fetch_add(o + 0, acc0, __ATOMIC_RELAXED, __HIP_MEMORY_SCOPE_AGENT);
            __hip_atomic_fetch_add(o + 1, acc1, __ATOMIC_RELAXED, __HIP_MEMORY_SCOPE_AGENT);
            acc0 = 0.0f; acc1 = 0.0f;
            cur = i;
        }

        // prefetch the streaming operand a few pairs ahead (global_prefetch_b8)
        if (p + 4 < p1) __builtin_prefetch(w_ij + (p + 4) * 64 + e, 0, 1);

        const float2 q = *(const float2*)(Q    + (long)i * 64 + e);
        const float2 k = *(const float2*)(Kt   + (long)j * 64 + e);
        const float2 v = *(const float2*)(Vt   + (long)j * 64 + e);
        const float2 w = *(const float2*)(w_ij + p * 64 + e);

        float part = q.x * w.x * k.x + q.y * w.y * k.y;
        // reduce across the 8 lanes of this head (wave32 shuffles)
        part += __shfl_xor(part, 1, 32);
        part += __shfl_xor(part, 2, 32);
        part += __shfl_xor(part, 4, 32);

        const float alpha = part * (pmask[p] * phi[p] * 0.25f);  // 1/sqrt(16)=0.25
        acc0 = fmaf(alpha, v.x, acc0);
        acc1 = fmaf(alpha, v.y, acc1);
    }

    float* o = out + (long)cur * 64 + e;
    __hip_atomic_fetch_add(o + 0, acc0, __ATOMIC_RELAXED, __HIP_MEMORY_SCOPE_AGENT);
    __hip_atomic_fetch_add(o + 1, acc1, __ATOMIC_RELAXED, __HIP_MEMORY_SCOPE_AGENT);
}

// ---------------------------------------------------------------------------
// Launch: inputs in setup_inputs() order:
//   0:x 1:w_ij 2:phi_r_cut 3:pair_mask 4:Wq 5:Wk 6:Wv 7:idx_i 8:idx_j
// Workspace: Q,K,V fp32 tables (3 * N * 64 * 4 bytes = 76.8 MB).
// ---------------------------------------------------------------------------
extern "C" void kernel_launch(void* const* d_in, const int* in_sizes, int n_in,
                              void* d_out, int out_size, void* d_ws, size_t ws_size,
                              hipStream_t stream) {
    const float* x     = (const float*)d_in[0];
    const float* w_ij  = (const float*)d_in[1];
    const float* phi   = (const float*)d_in[2];
    const float* pmask = (const float*)d_in[3];
    const float* Wq    = (const float*)d_in[4];
    const float* Wk    = (const float*)d_in[5];
    const float* Wv    = (const float*)d_in[6];
    const int*   idx_i = (const int*)d_in[7];
    const int*   idx_j = (const int*)d_in[8];
    float* out = (float*)d_out;

    const int  n_nodes = in_sizes[0] / 64;
    const long n_pairs = (long)in_sizes[2];

    float* Q = (float*)d_ws;
    float* K = Q + (size_t)n_nodes * 64;
    float* V = K + (size_t)n_nodes * 64;

    // Pass 1: WMMA projections (one wave per (16-node tile, head))
    {
        dim3 grid((n_nodes + 15) / 16);
        dim3 block(32, 4);
        proj16_wmma<<<grid, block, 0, stream>>>(x, Wq, Wk, Wv, Q, K, V, n_nodes);
    }

    // Pass 2: zero output
    {
        zero_f32<<<2048, 256, 0, stream>>>(out, (long)out_size);
    }

    // Pass 3: edge pass (64 pairs per wave, 8 waves per block)
    {
        const int  ppw    = 64;
        const long waves  = (n_pairs + ppw - 1) / ppw;
        const long blocks = (waves + 7) / 8;
        edge_attn<<<(int)blocks, 256, 0, stream>>>(Q, K, V, w_ij, phi, pmask,
                                                   idx_i, idx_j, out, n_pairs, ppw);
    }
}